// UserAttention_63376537420338
// MI455X (gfx1250) — compile-verified
//
#include <hip/hip_runtime.h>

// ---- Problem constants (from reference) ------------------------------------
#define B_N 2048
#define M_N 200
#define E_N 256
#define R_N 64

typedef float v2f __attribute__((ext_vector_type(2)));
typedef float v8f __attribute__((ext_vector_type(8)));
typedef int   v4i __attribute__((vector_size(16)));   // b128 payload type

#define AS1 __attribute__((address_space(1)))
#define AS3 __attribute__((address_space(3)))

// ---- CDNA5 async global->LDS copy (ASYNCcnt path), with safe fallback ------
#if defined(__has_builtin)
#if __has_builtin(__builtin_amdgcn_global_load_async_to_lds_b128)
#define HAVE_ASYNC_LDS 1
#endif
#endif
#ifndef HAVE_ASYNC_LDS
#define HAVE_ASYNC_LDS 0
#endif

__device__ __forceinline__ void cp16_async(float* lds_dst, const float* gsrc) {
#if HAVE_ASYNC_LDS
  __builtin_amdgcn_global_load_async_to_lds_b128(
      (AS1 v4i*)gsrc, (AS3 v4i*)lds_dst, 0, 0);
#else
  *(float4*)lds_dst = *(const float4*)gsrc;
#endif
}
__device__ __forceinline__ void wait_async_half() {
#if HAVE_ASYNC_LDS
  asm volatile("s_wait_asynccnt 25" ::: "memory");
#endif
}
__device__ __forceinline__ void wait_async_zero() {
#if HAVE_ASYNC_LDS
  asm volatile("s_wait_asynccnt 0" ::: "memory");
#endif
}

// ---- LDS layout (in floats) ------------------------------------------------
#define OFF_INTER 0
#define OFF_UW    (M_N * E_N)          // 51200
#define OFF_WF    (OFF_UW + E_N)       // 51456
#define OFF_ATTN  (OFF_WF + E_N)       // 51712 (256 slots, rows >=200 are 0)
#define OFF_RED   (OFF_ATTN + 256)     // 51968
#define SMEM_FLOATS (OFF_RED + 16)
#define SMEM_BYTES  (SMEM_FLOATS * 4)  // 207,936 B < 320 KB WGP LDS

__global__ __launch_bounds__(256)
void ua_fused_kernel(const float* __restrict__ user,
                     const float* __restrict__ item,
                     const float* __restrict__ inter,
                     const float* __restrict__ key_w,
                     const float* __restrict__ val_w,
                     float* __restrict__ out) {
  extern __shared__ float sm[];
  float* s_inter = sm + OFF_INTER;   // raw inter[b] : [M][E]
  float* s_uw    = sm + OFF_UW;      // uw[e] = item*wsum*user
  float* s_wf    = sm + OFF_WF;      // wf[e] = item*wsum
  float* s_attn  = sm + OFF_ATTN;    // scores -> attn
  float* s_red   = sm + OFF_RED;     // cross-wave reduction scratch

  const int t    = threadIdx.x;
  const int b    = blockIdx.x;
  const int wave = t >> 5;
  const int lane = t & 31;

  const float* gInter = inter + (size_t)b * (M_N * E_N);

  // ---- Issue async copies: half0 (rows 0..99) then half1 (rows 100..199) ---
  // 25 + 25 b128 instructions per wave; loads complete in issue order.
#pragma unroll
  for (int k = 0; k < 25; ++k) {
    const int f = k * 1024 + t * 4;                 // float index, 16B aligned
    cp16_async(s_inter + f, gInter + f);
  }
#pragma unroll
  for (int k = 0; k < 25; ++k) {
    const int f = 25600 + k * 1024 + t * 4;
    cp16_async(s_inter + f, gInter + f);
  }

  // ---- Per-column weights (overlaps with async DMA; uses LOADcnt) ----------
  float wsum = 0.f;
#pragma unroll 8
  for (int r = 0; r < R_N; ++r)
    wsum = fmaf(key_w[r * E_N + t], val_w[r * E_N + t], wsum);
  const float wf = wsum * item[(size_t)b * E_N + t];
  s_wf[t] = wf;
  s_uw[t] = wf * user[(size_t)b * E_N + t];

  // ---- Wait for half0 only; half1 keeps streaming during score compute -----
  wait_async_half();
  __syncthreads();

  const float4* uw4 = (const float4*)s_uw;
  const float4  u0 = uw4[lane];
  const float4  u1 = uw4[lane + 32];

  // scores[m] = sum_e inter[m][e] * uw[e]   (wave-per-row, 8 f32 per lane)
  for (int mm = wave; mm < 100; mm += 8) {
    const float4* row = (const float4*)(s_inter + mm * E_N);
    const float4 a0 = row[lane];
    const float4 a1 = row[lane + 32];
    float p = a0.x * u0.x + a0.y * u0.y + a0.z * u0.z + a0.w * u0.w
            + a1.x * u1.x + a1.y * u1.y + a1.z * u1.z + a1.w * u1.w;
#pragma unroll
    for (int off = 16; off; off >>= 1) p += __shfl_xor(p, off, 32);
    if (lane == 0) s_attn[mm] = p;
  }

  wait_async_zero();
  __syncthreads();

  for (int mm = wave; mm < 100; mm += 8) {
    const int m = 100 + mm;
    const float4* row = (const float4*)(s_inter + m * E_N);
    const float4 a0 = row[lane];
    const float4 a1 = row[lane + 32];
    float p = a0.x * u0.x + a0.y * u0.y + a0.z * u0.z + a0.w * u0.w
            + a1.x * u1.x + a1.y * u1.y + a1.z * u1.z + a1.w * u1.w;
#pragma unroll
    for (int off = 16; off; off >>= 1) p += __shfl_xor(p, off, 32);
    if (lane == 0) s_attn[m] = p;
  }
  __syncthreads();

  // ---- Softmax over m (block-wide max / sum) -------------------------------
  const float sval = (t < M_N) ? s_attn[t] : -__builtin_inff();
  float wm = sval;
#pragma unroll
  for (int off = 16; off; off >>= 1) wm = fmaxf(wm, __shfl_xor(wm, off, 32));
  if (lane == 0) s_red[wave] = wm;
  __syncthreads();
  float gmax = s_red[0];
#pragma unroll
  for (int i = 1; i < 8; ++i) gmax = fmaxf(gmax, s_red[i]);

  const float p = (t < M_N) ? __expf(sval - gmax) : 0.f;
  float ws = p;
#pragma unroll
  for (int off = 16; off; off >>= 1) ws += __shfl_xor(ws, off, 32);
  __syncthreads();                       // everyone done reading s_red (max)
  if (lane == 0) s_red[wave] = ws;
  __syncthreads();
  float gsum = 0.f;
#pragma unroll
  for (int i = 0; i < 8; ++i) gsum += s_red[i];
  s_attn[t] = p / gsum;                  // t>=200 slots become 0 (harmless)
  __syncthreads();

  // ---- out[j] = wf[j] * sum_m attn[m]*inter[m][j]  via V_WMMA_F32_16X16X4 --
  // A (16x4): attn replicated on all 16 rows -> every D row equals out tile.
  //   lanes 0-15 : {attn[kb+0], attn[kb+1]}, lanes 16-31: {attn[kb+2], attn[kb+3]}
  // B (4x16): v0 = rows K0(lanes0-15)/K1(lanes16-31), v1 = rows K2/K3.
  const int ln = lane & 15;
  const int hi = lane >> 4;
#pragma unroll
  for (int tile = wave; tile < 16; tile += 8) {      // 2 n-tiles per wave
    const int n0 = tile * 16;
    v8f c = {0.f, 0.f, 0.f, 0.f, 0.f, 0.f, 0.f, 0.f};
    for (int kb = 0; kb < M_N; kb += 4) {            // 50 K-steps
      v2f A;
      A[0] = s_attn[kb + 2 * hi];
      A[1] = s_attn[kb + 2 * hi + 1];
      v2f Bm;
      Bm[0] = s_inter[(kb + hi) * E_N + n0 + ln];
      Bm[1] = s_inter[(kb + 2 + hi) * E_N + n0 + ln];
      c = __builtin_amdgcn_wmma_f32_16x16x4_f32(
          /*neg_a=*/false, A, /*neg_b=*/false, Bm,
          /*c_mod=*/(short)0, c, /*reuse_a=*/false, /*reuse_b=*/false);
    }
    // D row 0 (c[0], lanes 0-15) carries out[n0+ln]; rows are identical.
    if (hi == 0) out[(size_t)b * E_N + n0 + ln] = s_wf[n0 + ln] * c[0];
  }
}

// ---- Host entry ------------------------------------------------------------
extern "C" void kernel_launch(void* const* d_in, const int* in_sizes, int n_in,
                              void* d_out, int out_size, void* d_ws, size_t ws_size,
                              hipStream_t stream) {
  (void)in_sizes; (void)n_in; (void)out_size; (void)d_ws; (void)ws_size;
  const float* user  = (const float*)d_in[0];
  const float* item  = (const float*)d_in[1];
  const float* inter = (const float*)d_in[2];
  const float* key_w = (const float*)d_in[3];
  const float* val_w = (const float*)d_in[4];
  float* out = (float*)d_out;

  // Raise the dynamic-LDS cap (208 KB; WGP LDS is 320 KB). Not a stream op,
  // deterministic, safe under graph capture.
  (void)hipFuncSetAttribute((const void*)ua_fused_kernel,
                            hipFuncAttributeMaxDynamicSharedMemorySize,
                            SMEM_BYTES);

  ua_fused_kernel<<<B_N, 256, SMEM_BYTES, stream>>>(user, item, inter,
                                                    key_w, val_w, out);
}